// SIOGNN_26508538151164
// MI455X (gfx1250) — compile-verified
//
#include <hip/hip_runtime.h>
#include <hip/hip_bf16.h>
#include <math.h>

#define LRELU_A 0.2f
#define NEGV   (-9.0e15f)
#define Bq 64
#define Sq 50
#define Gq 50
#define Kq 12
#define Dq 256

typedef float v2f __attribute__((ext_vector_type(2)));
typedef float v8f __attribute__((ext_vector_type(8)));
typedef int   v4i __attribute__((ext_vector_type(4)));

#if __has_builtin(__builtin_amdgcn_global_load_async_to_lds_b128) && \
    __has_builtin(__builtin_amdgcn_s_wait_asynccnt)
#define HAVE_ASYNC_LDS 1
#else
#define HAVE_ASYNC_LDS 0
#endif

__device__ __forceinline__ float lrelu(float x) { return x >= 0.f ? x : LRELU_A * x; }

// ---------- item_for_node[b, alias[b,p]] = cur[b,p] ----------
__global__ void k_item_for_node(const int* __restrict__ alias, const int* __restrict__ cur,
                                int* __restrict__ ifn) {
    int b = blockIdx.x, p = threadIdx.x;
    ifn[b * Gq + alias[b * Gq + p]] = cur[b * Gq + p];
}

// ---------- out[j, 0:256] (stride ld) = table[idx[j], 0:256] ----------
__global__ void k_gather(const int* __restrict__ idx, const float* __restrict__ table,
                         float* __restrict__ out, int ld) {
    int j = blockIdx.x, t = threadIdx.x;  // 64 threads, one float4 each
    const float4* src = (const float4*)(table + (size_t)idx[j] * Dq);
    float4* dst = (float4*)(out + (size_t)j * ld);
    dst[t] = src[t];
}

// ---------- dst[j, 0:256] (stride ld) = src[j, 0:256] (stride 256) ----------
__global__ void k_copy256(const float* __restrict__ src, float* __restrict__ dst, int ld) {
    int j = blockIdx.x, t = threadIdx.x;  // 256 threads
    dst[(size_t)j * ld + t] = src[(size_t)j * Dq + t];
}

// ---------- dst[j, 0:256] (stride ld) = pos[j%S, 0:256] ----------
__global__ void k_pos_fill(const float* __restrict__ pos, float* __restrict__ dst, int ld) {
    int j = blockIdx.x, t = threadIdx.x;
    dst[(size_t)j * ld + t] = pos[(j % Sq) * Dq + t];
}

// ---------- tmp[b,s,:] = sum_t path[b,s,t] * hidden[b,t,:] ----------
__global__ void k_pathmm(const float* __restrict__ path, const float* __restrict__ hidden,
                         float* __restrict__ out) {
    int j = blockIdx.x, t = threadIdx.x;  // 256 threads
    int b = j / Sq, s = j % Sq;
    const float* prow = path + ((size_t)b * Sq + s) * Sq;
    const float* hb = hidden + (size_t)b * Sq * Dq + t;
    float acc = 0.f;
    for (int tt = 0; tt < Sq; ++tt) acc += prow[tt] * hb[(size_t)tt * Dq];
    out[(size_t)j * Dq + t] = acc;
}

// ---------- ragged top-K neighbor sampling ----------
__global__ void k_neighbors(const int* __restrict__ items, int M,
                            const int* __restrict__ cur, const int* __restrict__ alias,
                            const float* __restrict__ adj, const int* __restrict__ ifn,
                            int* __restrict__ nb_out, float* __restrict__ nw_out) {
    int tid = blockIdx.x * blockDim.x + threadIdx.x;
    if (tid >= Bq * M) return;
    int b = tid / M;
    int item = items[(size_t)tid];
    bool valid = item > 0;
    int ind = 0;
    for (int g = 0; g < Gq; ++g) {
        if (cur[b * Gq + g] == item) { ind = g; break; }
    }
    int node = alias[b * Gq + ind];
    const float* row = adj + ((size_t)b * Gq + node) * Gq;
    float w[Gq];
    for (int g = 0; g < Gq; ++g) { float v = row[g]; w[g] = v > 0.f ? v : NEGV; }
    for (int kk = 0; kk < Kq; ++kk) {
        int bi = 0; float bv = w[0];
        for (int g = 1; g < Gq; ++g) if (w[g] > bv) { bv = w[g]; bi = g; }  // first index on ties
        w[bi] = -__builtin_inff();
        bool ok = (bv > NEGV * 0.5f) && valid;
        nb_out[(size_t)tid * Kq + kk] = ok ? ifn[b * Gq + bi] : 0;
        nw_out[(size_t)tid * Kq + kk] = ok ? bv : NEGV;
    }
}

// ---------- softmax-over-K weighted neighbor aggregation ----------
// use_idx=1: neighbor rows gathered from src via nb; use_idx=0: dense rows j*K+k
__global__ void k_agg(const int* __restrict__ nb, const float* __restrict__ nw,
                      const float* __restrict__ src, int use_idx,
                      float* __restrict__ out, int ld) {
    int j = blockIdx.x, t = threadIdx.x;  // 64 threads, one float4 lane each
    float lg[Kq];
    float mx = -__builtin_inff();
    for (int k = 0; k < Kq; ++k) {
        int n = nb[(size_t)j * Kq + k];
        float v = (n > 0) ? nw[(size_t)j * Kq + k] : NEGV;
        lg[k] = v; mx = fmaxf(mx, v);
    }
    float z = 0.f;
    for (int k = 0; k < Kq; ++k) { lg[k] = __expf(lg[k] - mx); z += lg[k]; }
    float inv = 1.f / z;
    float4 acc = make_float4(0.f, 0.f, 0.f, 0.f);
    for (int k = 0; k < Kq; ++k) {
        size_t r = use_idx ? (size_t)nb[(size_t)j * Kq + k] : ((size_t)j * Kq + k);
        float4 x = *((const float4*)(src + r * Dq) + t);
        float p = lg[k] * inv;
        acc.x += p * x.x; acc.y += p * x.y; acc.z += p * x.z; acc.w += p * x.w;
    }
    *((float4*)(out + (size_t)j * ld) + t) = acc;
}

// ---------- ht[b,:] = h[b, last_b, :] ----------
__global__ void k_ht(const float* __restrict__ mask, const float* __restrict__ h,
                     float* __restrict__ ht) {
    int b = blockIdx.x, t = threadIdx.x;
    float s = 0.f;
    for (int i = 0; i < Sq; ++i) s += mask[b * Sq + i];
    int last = (int)(s + 0.5f) - 1;
    ht[(size_t)b * Dq + t] = h[((size_t)b * Sq + last) * Dq + t];
}

// ---------- e[j] = sum_d lrelu(xw[j,d] + yw[b,d] + bias[d]) * qv[d], optional mask ----------
__global__ void k_score(const float* __restrict__ xw, const float* __restrict__ yw,
                        const float* __restrict__ bias, const float* __restrict__ qv,
                        const float* __restrict__ mask, float* __restrict__ out) {
    __shared__ float red[256];
    int j = blockIdx.x, t = threadIdx.x;
    int b = j / Sq;
    red[t] = lrelu(xw[(size_t)j * Dq + t] + yw[(size_t)b * Dq + t] + bias[t]) * qv[t];
    __syncthreads();
    for (int s = 128; s > 0; s >>= 1) {
        if (t < s) red[t] += red[t + s];
        __syncthreads();
    }
    if (t == 0) {
        float e = red[0];
        if (mask) e = (mask[j] > 0.f) ? e : NEGV;
        out[j] = e;
    }
}

// ---------- out[b,:] = sum_s softmax(e[b,:])[s] * H[b,s,:] ----------
__global__ void k_smsum(const float* __restrict__ e, const float* __restrict__ H,
                        float* __restrict__ out) {
    int b = blockIdx.x, t = threadIdx.x;  // 256 threads
    float mx = -__builtin_inff();
    for (int s = 0; s < Sq; ++s) mx = fmaxf(mx, e[b * Sq + s]);
    float z = 0.f;
    for (int s = 0; s < Sq; ++s) z += __expf(e[b * Sq + s] - mx);
    float inv = 1.f / z, acc = 0.f;
    for (int s = 0; s < Sq; ++s)
        acc += __expf(e[b * Sq + s] - mx) * H[((size_t)b * Sq + s) * Dq + t];
    out[(size_t)b * Dq + t] = acc * inv;
}

// ---------- C[M,256] = act(A[M,Kd] @ W[Kd,256] + bias) via V_WMMA_F32_16X16X4_F32 ----------
// 4 waves / block; block computes a 128(M) x 32(N) stripe.
// B panel [Kd x 32] staged in LDS (async global->LDS when available), shared by all waves.
// Each wave: 2x2 register-blocked (32x32), 4 WMMAs per k-step.
__global__ void k_wmma_gemm2(const float* __restrict__ A, const float* __restrict__ W,
                             const float* __restrict__ bias, float* __restrict__ C,
                             int Kd, int lda, int ldc, int act, int Mtot) {
    extern __shared__ float Bs[];  // Kd * 32 floats
    const int tid = threadIdx.x;   // 128 threads = 4 waves
    const int n0 = blockIdx.x * 32;

    // ---- stage B panel into LDS ----
    for (int i = tid; i < Kd * 8; i += 128) {     // float4 granules
        int k = i >> 3, c = (i & 7) << 2;
#if HAVE_ASYNC_LDS
        __builtin_amdgcn_global_load_async_to_lds_b128(
            (__attribute__((address_space(1))) v4i*)(W + (size_t)k * 256 + n0 + c),
            (__attribute__((address_space(3))) v4i*)(Bs + (size_t)k * 32 + c), 0, 0);
#else
        *(float4*)(Bs + (size_t)k * 32 + c) = *(const float4*)(W + (size_t)k * 256 + n0 + c);
#endif
    }
#if HAVE_ASYNC_LDS
    __builtin_amdgcn_s_wait_asynccnt(0);
#endif
    __syncthreads();

    const int lane = tid & 31, wv = tid >> 5;
    const int rlo = lane & 15, hi = lane >> 4;
    const int m0 = blockIdx.y * 128 + wv * 32;
    if (m0 >= Mtot) return;

    const float* a0p = A + (size_t)(m0 + rlo) * (size_t)lda;
    const float* a1p = A + (size_t)(m0 + 16 + rlo) * (size_t)lda;
    const float* b0p = Bs + rlo;
    const float* b1p = Bs + 16 + rlo;

    v8f acc00 = {0.f,0.f,0.f,0.f,0.f,0.f,0.f,0.f};
    v8f acc01 = {0.f,0.f,0.f,0.f,0.f,0.f,0.f,0.f};
    v8f acc10 = {0.f,0.f,0.f,0.f,0.f,0.f,0.f,0.f};
    v8f acc11 = {0.f,0.f,0.f,0.f,0.f,0.f,0.f,0.f};
    for (int k0 = 0; k0 < Kd; k0 += 4) {
        int ka = k0 + 2 * hi;
        v2f a0; a0.x = a0p[ka];          a0.y = a0p[ka + 1];
        v2f a1; a1.x = a1p[ka];          a1.y = a1p[ka + 1];
        v2f b0; b0.x = b0p[ka * 32];     b0.y = b0p[(ka + 1) * 32];
        v2f b1; b1.x = b1p[ka * 32];     b1.y = b1p[(ka + 1) * 32];
        acc00 = __builtin_amdgcn_wmma_f32_16x16x4_f32(false, a0, false, b0, (short)0, acc00, false, false);
        acc01 = __builtin_amdgcn_wmma_f32_16x16x4_f32(false, a0, false, b1, (short)0, acc01, false, false);
        acc10 = __builtin_amdgcn_wmma_f32_16x16x4_f32(false, a1, false, b0, (short)0, acc10, false, false);
        acc11 = __builtin_amdgcn_wmma_f32_16x16x4_f32(false, a1, false, b1, (short)0, acc11, false, false);
    }

    float bb0 = bias ? bias[n0 + rlo] : 0.f;
    float bb1 = bias ? bias[n0 + 16 + rlo] : 0.f;
    for (int r = 0; r < 8; ++r) {
        size_t mr0 = (size_t)m0 + r + 8 * hi;
        size_t mr1 = mr0 + 16;
        float v00 = acc00[r] + bb0, v01 = acc01[r] + bb1;
        float v10 = acc10[r] + bb0, v11 = acc11[r] + bb1;
        if (act) { v00 = lrelu(v00); v01 = lrelu(v01); v10 = lrelu(v10); v11 = lrelu(v11); }
        C[mr0 * (size_t)ldc + n0 + rlo]      = v00;
        C[mr0 * (size_t)ldc + n0 + 16 + rlo] = v01;
        C[mr1 * (size_t)ldc + n0 + rlo]      = v10;
        C[mr1 * (size_t)ldc + n0 + 16 + rlo] = v11;
    }
}

extern "C" void kernel_launch(void* const* d_in, const int* in_sizes, int n_in,
                              void* d_out, int out_size, void* d_ws, size_t ws_size,
                              hipStream_t stream) {
    (void)in_sizes; (void)n_in; (void)out_size; (void)ws_size;

    const int*   data  = (const int*)d_in[0];
    const float* mask  = (const float*)d_in[1];
    const float* adj   = (const float*)d_in[2];
    const int*   alias = (const int*)d_in[3];
    const int*   cur   = (const int*)d_in[4];
    const float* path  = (const float*)d_in[5];
    const float* emb   = (const float*)d_in[6];
    const float* pos   = (const float*)d_in[7];
    const float* w_3   = (const float*)d_in[8];
    const float* b_1   = (const float*)d_in[9];
    const float* qv    = (const float*)d_in[10];
    const float* w_g1  = (const float*)d_in[11];
    const float* w_g2  = (const float*)d_in[12];
    const float* b_2   = (const float*)d_in[13];
    const float* w_h   = (const float*)d_in[14];
    const float* w_p   = (const float*)d_in[15];
    const float* w_1   = (const float*)d_in[16];
    const float* w_2   = (const float*)d_in[17];
    const float* b_l   = (const float*)d_in[18];
    const float* q_l   = (const float*)d_in[19];
    const float* w_agg0 = (const float*)d_in[20];
    const float* w_agg1 = w_agg0 + (size_t)512 * 256;
    const float* b_agg0 = (const float*)d_in[21];
    const float* b_agg1 = b_agg0 + 256;

    const int M0 = Bq * Sq;        // 3200
    const int M1 = Bq * Sq * Kq;   // 38400
    const int M2 = M1 * Kq;        // 460800

    char* wp = (char*)d_ws;
    auto alloc = [&](size_t bytes) -> void* {
        void* p = (void*)wp;
        wp += (bytes + 255) & ~(size_t)255;
        return p;
    };
    int*   ifn     = (int*)alloc((size_t)Bq * Gq * 4);
    float* hidden  = (float*)alloc((size_t)M0 * Dq * 4);
    float* tmp     = (float*)alloc((size_t)M0 * Dq * 4);
    float* h_loc   = (float*)alloc((size_t)M0 * Dq * 4);
    int*   n1_nb   = (int*)alloc((size_t)M1 * 4);
    float* n1_w    = (float*)alloc((size_t)M1 * 4);
    int*   n2_nb   = (int*)alloc((size_t)M2 * 4);
    float* n2_w    = (float*)alloc((size_t)M2 * 4);
    float* cat1    = (float*)alloc((size_t)M1 * 512 * 4);
    float* nev1    = (float*)alloc((size_t)M1 * Dq * 4);
    float* cat0    = (float*)alloc((size_t)M0 * 512 * 4);
    float* nev0    = (float*)alloc((size_t)M0 * Dq * 4);
    float* cat0b   = (float*)alloc((size_t)M0 * 512 * 4);
    float* h_glob  = (float*)alloc((size_t)M0 * Dq * 4);
    float* cat_s   = (float*)alloc((size_t)M0 * 512 * 4);
    float* s_buf   = (float*)alloc((size_t)M0 * Dq * 4);
    float* hw1     = (float*)alloc((size_t)M0 * Dq * 4);
    float* htw2    = (float*)alloc((size_t)Bq * Dq * 4);
    float* ht      = (float*)alloc((size_t)Bq * Dq * 4);
    float* e_buf   = (float*)alloc((size_t)M0 * 4);
    float* s_l     = (float*)alloc((size_t)Bq * Dq * 4);
    float* sw      = (float*)alloc((size_t)M0 * Dq * 4);
    float* slw     = (float*)alloc((size_t)Bq * Dq * 4);
    float* beta    = (float*)alloc((size_t)M0 * 4);
    float* s_g     = (float*)alloc((size_t)Bq * Dq * 4);
    float* cat_f   = (float*)alloc((size_t)Bq * 512 * 4);

    auto gemm = [&](const float* A, const float* W, const float* bias, float* C,
                    int Kd, int lda, int act, int M) {
        dim3 grid(8, (M + 127) / 128);
        size_t shmem = (size_t)Kd * 32 * sizeof(float);
        k_wmma_gemm2<<<grid, 128, shmem, stream>>>(A, W, bias, C, Kd, lda, 256, act, M);
    };

    // ---- graph bookkeeping + local intent front-end ----
    k_item_for_node<<<Bq, Gq, 0, stream>>>(alias, cur, ifn);
    k_gather<<<M0, 64, 0, stream>>>(data, emb, hidden, Dq);
    k_pathmm<<<M0, 256, 0, stream>>>(path, hidden, tmp);
    gemm(tmp, w_p, nullptr, h_loc, 256, 256, 1, M0);

    // ---- neighbor sampling (2 hops) ----
    k_neighbors<<<(Bq * Sq + 127) / 128, 128, 0, stream>>>(data, Sq, cur, alias, adj, ifn, n1_nb, n1_w);
    k_neighbors<<<(M1 + 127) / 128, 128, 0, stream>>>(n1_nb, Sq * Kq, cur, alias, adj, ifn, n2_nb, n2_w);

    // ---- GlobalIntent layer 0 ----
    k_gather<<<M1, 64, 0, stream>>>(n1_nb, emb, cat1, 512);
    k_gather<<<M0, 64, 0, stream>>>(data, emb, cat0, 512);
    k_agg<<<M1, 64, 0, stream>>>(n2_nb, n2_w, emb, 1, cat1 + 256, 512);
    k_agg<<<M0, 64, 0, stream>>>(n1_nb, n1_w, emb, 1, cat0 + 256, 512);
    gemm(cat1, w_agg0, b_agg0, nev1, 512, 512, 1, M1);
    gemm(cat0, w_agg0, b_agg0, nev0, 512, 512, 1, M0);

    // ---- GlobalIntent layer 1 ----
    k_copy256<<<M0, 256, 0, stream>>>(nev0, cat0b, 512);
    k_agg<<<M0, 64, 0, stream>>>(n1_nb, n1_w, nev1, 0, cat0b + 256, 512);
    gemm(cat0b, w_agg1, b_agg1, h_glob, 512, 512, 1, M0);

    // ---- fusion: s = lrelu([pos | h_global] @ w_3 + b_1) ----
    k_pos_fill<<<M0, 256, 0, stream>>>(pos, cat_s, 512);
    k_copy256<<<M0, 256, 0, stream>>>(h_glob, cat_s + 256, 512);
    gemm(cat_s, w_3, b_1, s_buf, 512, 512, 1, M0);

    // ---- local intent attention ----
    k_ht<<<Bq, 256, 0, stream>>>(mask, h_loc, ht);
    gemm(h_loc, w_1, nullptr, hw1, 256, 256, 0, M0);
    gemm(ht, w_2, nullptr, htw2, 256, 256, 0, Bq);
    k_score<<<M0, 256, 0, stream>>>(hw1, htw2, b_l, q_l, mask, e_buf);
    k_smsum<<<Bq, 256, 0, stream>>>(e_buf, h_loc, s_l);

    // ---- global attention ----
    gemm(s_buf, w_g1, nullptr, sw, 256, 256, 0, M0);
    gemm(s_l, w_g2, nullptr, slw, 256, 256, 0, Bq);
    k_score<<<M0, 256, 0, stream>>>(sw, slw, b_2, qv, nullptr, beta);
    k_smsum<<<Bq, 256, 0, stream>>>(beta, s_buf, s_g);

    // ---- output: lrelu([s_l | s_g] @ w_h) ----
    k_copy256<<<Bq, 256, 0, stream>>>(s_l, cat_f, 512);
    k_copy256<<<Bq, 256, 0, stream>>>(s_g, cat_f + 256, 512);
    gemm(cat_f, w_h, nullptr, (float*)d_out, 512, 512, 1, Bq);
}